// MotionMemory_72112500899924
// MI455X (gfx1250) — compile-verified
//
#include <hip/hip_runtime.h>

typedef __attribute__((ext_vector_type(2))) float v2f;
typedef __attribute__((ext_vector_type(8))) float v8f;

#define T_TOT   32768      // 8*4096 tokens
#define FEAT    2048
#define MSLOT   10
#define MPAD    16
#define TPB     128        // tokens per block in main kernel
#define EPSN    1e-12f

// ws layout (float units):
//   [0, 16*FEAT)                    : mn  = l2norm(m_items), rows 10..15 zeroed
//   [16*FEAT, 16*FEAT+10*FEAT)      : g_upd accumulator (10 x 2048)
//   then 16 unsigned                : g_cmax order-preserving float keys

__device__ __forceinline__ unsigned fkey(float x) {
    unsigned b = __float_as_uint(x);
    return (b & 0x80000000u) ? ~b : (b | 0x80000000u);
}
__device__ __forceinline__ float fkey_inv(unsigned k) {
    unsigned b = (k & 0x80000000u) ? (k & 0x7fffffffu) : ~k;
    return __uint_as_float(b);
}

__global__ __launch_bounds__(256) void mm_prep(const float* __restrict__ m_items,
                                               float* __restrict__ ws) {
    float* mn = ws;
    float* g_upd = ws + MPAD * FEAT;
    unsigned* g_cmax = (unsigned*)(ws + MPAD * FEAT + MSLOT * FEAT);
    int b = blockIdx.x, tid = threadIdx.x;
    __shared__ float red[8];
    if (b < MSLOT) {
        float v[8]; float ss = 0.f;
        #pragma unroll
        for (int i = 0; i < 8; ++i) {
            v[i] = m_items[b * FEAT + tid + i * 256];
            ss += v[i] * v[i];
        }
        #pragma unroll
        for (int off = 16; off; off >>= 1) ss += __shfl_xor(ss, off, 32);
        if ((tid & 31) == 0) red[tid >> 5] = ss;
        __syncthreads();
        if (tid == 0) { float t = 0.f; for (int i = 0; i < 8; ++i) t += red[i]; red[0] = t; }
        __syncthreads();
        float inv = 1.0f / fmaxf(sqrtf(red[0]), EPSN);
        #pragma unroll
        for (int i = 0; i < 8; ++i) mn[b * FEAT + tid + i * 256] = v[i] * inv;
    } else if (b == MSLOT) {
        for (int i = tid; i < (MPAD - MSLOT) * FEAT; i += 256) mn[MSLOT * FEAT + i] = 0.f;
    } else {
        for (int i = tid; i < MSLOT * FEAT; i += 256) g_upd[i] = 0.f;
        if (tid < 16) g_cmax[tid] = 0u;   // below fkey(-1.0) so any sim wins
    }
}

__global__ __launch_bounds__(256) void mm_main(const float* __restrict__ feat,
                                               const float* __restrict__ m_items,
                                               float* __restrict__ ws,
                                               float* __restrict__ out) {
    const float* mn = ws;
    float* g_upd = ws + MPAD * FEAT;
    unsigned* g_cmax = (unsigned*)(ws + MPAD * FEAT + MSLOT * FEAT);

    __shared__ float s_tok[TPB][12];       // [0..9]=softmax p, [10]=exp(rowmax)=w, [11]=kidx bits
    __shared__ unsigned s_cmax[16];

    const int tid  = threadIdx.x;
    const int lane = tid & 31, wave = tid >> 5;
    const int half = lane >> 4, n = lane & 15;   // n: A-row (token-in-tile) and B-col (slot)
    const size_t tok0 = (size_t)blockIdx.x * TPB;

    if (tid < 16) s_cmax[tid] = 0u;

    // ---------- Phase 1: sim tile = (f @ mn^T) via V_WMMA_F32_16X16X4_F32, fused ||f||^2 ----------
    const size_t tileTok = tok0 + (size_t)wave * 16;
    const float* pA = feat + (tileTok + n) * FEAT + 2 * half;   // A 16x4: lane row=n, k = 2*half + v
    const float* pB = mn + (size_t)n * FEAT + 2 * half;         // B 4x16: lane col=n, k = 2*half + v (rows>=10 zero)
    v8f c = {};
    float nrm = 0.f;
    #pragma unroll 4
    for (int k0 = 0; k0 < FEAT; k0 += 4) {
        v2f a = *(const v2f*)(pA + k0);
        v2f b = *(const v2f*)(pB + k0);
        nrm += a.x * a.x + a.y * a.y;
        c = __builtin_amdgcn_wmma_f32_16x16x4_f32(false, a, false, b, (short)0, c, false, false);
    }
    nrm += __shfl_xor(nrm, 16, 32);        // lanes n and n+16 both covered k%4 halves -> full ||f||^2

    __syncthreads();                        // s_cmax init visible before LDS atomics

    // ---------- Phase 1b: per-row softmax / argmax / colmax on the striped D fragment ----------
    float cmax = -3.0e38f;
    #pragma unroll
    for (int r = 0; r < 8; ++r) {
        int row = r + 8 * half;                       // D: lane,vgpr r -> M = r + 8*half, N = n
        float n2  = __shfl(nrm, row, 32);             // ||f||^2 of this row lives on lane 'row'
        float inv = 1.0f / fmaxf(sqrtf(n2), EPSN);
        float sm  = (n < MSLOT) ? c[r] * inv : -3.0e38f;
        cmax = fmaxf(cmax, sm);
        float rmax = sm;
        #pragma unroll
        for (int off = 8; off; off >>= 1) rmax = fmaxf(rmax, __shfl_xor(rmax, off, 32));
        unsigned long long bal = __ballot(sm == rmax);
        unsigned mymask = (unsigned)(bal >> (16 * half)) & 0xffffu;
        int kidx = __ffs(mymask) - 1;                 // first max index == jnp.argmax tie-break
        float e = (n < MSLOT) ? __expf(sm - rmax) : 0.f;
        float rsum = e;
        #pragma unroll
        for (int off = 8; off; off >>= 1) rsum += __shfl_xor(rsum, off, 32);
        float p = e / rsum;
        int tl = wave * 16 + row;
        if (n < MSLOT) s_tok[tl][n] = p;
        if (n == 0) {
            s_tok[tl][10] = __expf(rmax);             // w = exp(sim[t,kidx]); colmax scaling deferred
            s_tok[tl][11] = __int_as_float(kidx);
        }
    }
    cmax = fmaxf(cmax, __shfl_xor(cmax, 16, 32));     // merge row halves
    if (lane < MSLOT) atomicMax(&s_cmax[lane], fkey(cmax));
    __syncthreads();
    if (tid < MSLOT) atomicMax(&g_cmax[tid], s_cmax[tid]);

    // ---------- Phase 2: stream feat_update + per-block VQ accumulation ----------
    #pragma unroll 1
    for (int pass = 0; pass < 2; ++pass) {
        const int j = pass * 1024 + tid * 4;          // thread owns 4 contiguous features
        float4 mit[MSLOT];
        #pragma unroll
        for (int m = 0; m < MSLOT; ++m)
            mit[m] = *(const float4*)(m_items + m * FEAT + j);
        float4 acc[MSLOT];
        #pragma unroll
        for (int m = 0; m < MSLOT; ++m) acc[m] = make_float4(0.f, 0.f, 0.f, 0.f);

        for (int t = 0; t < TPB; ++t) {
            const size_t tok = tok0 + t;
            __builtin_prefetch(feat + (tok + 8) * FEAT + j, 0, 0);
            float4 val = *(const float4*)(feat + tok * FEAT + j);
            float4 p0 = *(const float4*)&s_tok[t][0];
            float4 p1 = *(const float4*)&s_tok[t][4];
            float4 p2 = *(const float4*)&s_tok[t][8];
            const int   kt = __float_as_int(p2.w);
            const float w  = p2.z;
            const float pr[MSLOT] = {p0.x, p0.y, p0.z, p0.w, p1.x, p1.y, p1.z, p1.w, p2.x, p2.y};
            float4 o = val;
            #pragma unroll
            for (int m = 0; m < MSLOT; ++m) {
                o.x = fmaf(pr[m], mit[m].x, o.x);
                o.y = fmaf(pr[m], mit[m].y, o.y);
                o.z = fmaf(pr[m], mit[m].z, o.z);
                o.w = fmaf(pr[m], mit[m].w, o.w);
                float sel = (kt == m) ? w : 0.f;      // hard-assignment mask * w
                acc[m].x = fmaf(sel, val.x, acc[m].x);
                acc[m].y = fmaf(sel, val.y, acc[m].y);
                acc[m].z = fmaf(sel, val.z, acc[m].z);
                acc[m].w = fmaf(sel, val.w, acc[m].w);
            }
            *(float4*)(out + tok * FEAT + j) = o;
        }
        #pragma unroll
        for (int m = 0; m < MSLOT; ++m) {
            if (acc[m].x != 0.f || acc[m].y != 0.f || acc[m].z != 0.f || acc[m].w != 0.f) {
                float* gp = g_upd + m * FEAT + j;
                atomicAdd(gp + 0, acc[m].x);
                atomicAdd(gp + 1, acc[m].y);
                atomicAdd(gp + 2, acc[m].z);
                atomicAdd(gp + 3, acc[m].w);
            }
        }
    }
}

__global__ __launch_bounds__(256) void mm_finalize(const float* __restrict__ m_items,
                                                   const float* __restrict__ ws,
                                                   float* __restrict__ out) {
    const float* g_upd = ws + MPAD * FEAT;
    const unsigned* g_cmax = (const unsigned*)(ws + MPAD * FEAT + MSLOT * FEAT);
    const int m = blockIdx.x, tid = threadIdx.x;
    __shared__ float red[8];
    const float scale = __expf(-fkey_inv(g_cmax[m]));   // w normalization: / exp(colmax[m])
    float v[8]; float ss = 0.f;
    #pragma unroll
    for (int i = 0; i < 8; ++i) {
        int j = tid + i * 256;
        v[i] = g_upd[m * FEAT + j] * scale + m_items[m * FEAT + j];
        ss += v[i] * v[i];
    }
    #pragma unroll
    for (int off = 16; off; off >>= 1) ss += __shfl_xor(ss, off, 32);
    if ((tid & 31) == 0) red[tid >> 5] = ss;
    __syncthreads();
    if (tid == 0) { float t = 0.f; for (int i = 0; i < 8; ++i) t += red[i]; red[0] = t; }
    __syncthreads();
    const float inv = 1.0f / fmaxf(sqrtf(red[0]), EPSN);
    float* o = out + (size_t)T_TOT * FEAT + m * FEAT;
    #pragma unroll
    for (int i = 0; i < 8; ++i) o[tid + i * 256] = v[i] * inv;
}

extern "C" void kernel_launch(void* const* d_in, const int* in_sizes, int n_in,
                              void* d_out, int out_size, void* d_ws, size_t ws_size,
                              hipStream_t stream) {
    (void)in_sizes; (void)n_in; (void)out_size; (void)ws_size;
    const float* feat    = (const float*)d_in[0];   // [8,4096,2048] f32
    const float* m_items = (const float*)d_in[1];   // [10,2048]    f32
    float* out = (float*)d_out;                     // feat_update (67108864) ++ mem_update (20480)
    float* ws  = (float*)d_ws;                      // ~213 KB used

    mm_prep<<<MSLOT + 2, 256, 0, stream>>>(m_items, ws);
    mm_main<<<T_TOT / TPB, 256, 0, stream>>>(feat, m_items, ws, out);
    mm_finalize<<<MSLOT, 256, 0, stream>>>(m_items, ws, out);
}